// GCNNet_69947837383269
// MI455X (gfx1250) — compile-verified
//
#include <hip/hip_runtime.h>
#include <hip/hip_bf16.h>

#define NFEAT 128
#define NHID  64
#define NCLS  40
#define NCLS_PAD 48   // 3 column tiles of 16

typedef float v2f __attribute__((ext_vector_type(2)));
typedef float v8f __attribute__((ext_vector_type(8)));

// ---------------- degree / norm ----------------

__global__ __launch_bounds__(256) void k_fill1(float* __restrict__ deg, int n) {
  int i = blockIdx.x * blockDim.x + threadIdx.x;
  if (i < n) deg[i] = 1.0f;   // self-loop contribution
}

__global__ __launch_bounds__(256) void k_deg_accum(const int* __restrict__ col,
                                                   float* __restrict__ deg, int E) {
  int e = blockIdx.x * blockDim.x + threadIdx.x;
  if (e < E) atomicAdd(&deg[col[e]], 1.0f);
}

__global__ __launch_bounds__(256) void k_rsqrt(float* __restrict__ d, int n) {
  int i = blockIdx.x * blockDim.x + threadIdx.x;
  if (i < n) d[i] = rsqrtf(d[i]);   // deg >= 1 always (self-loops)
}

// ---------------- weight prep: pad + transpose ----------------
// Bt[n * K + k] = (n < Ncols) ? B[k * Ncols + n] : 0   for n in [0, NPad), k in [0, K)
__global__ __launch_bounds__(256) void k_pad_transpose(const float* __restrict__ B,
                                                       float* __restrict__ Bt,
                                                       int K, int Ncols, int NPad) {
  int i = blockIdx.x * blockDim.x + threadIdx.x;   // over NPad * K
  if (i >= NPad * K) return;
  int n = i / K;
  int k = i - n * K;
  Bt[i] = (n < Ncols) ? B[k * Ncols + n] : 0.0f;
}

// ---------------- fp32 WMMA GEMM: C[M x Ncols] = A[M x K] @ B[K x Ncols] ----------------
// B supplied transposed+padded as Bt[NPad x K]. One wave = one 16x16 C tile,
// V_WMMA_F32_16X16X4_F32 stepping K by 4. M multiple of 16 (100000 = 6250*16).
// Branch-free inner loop: one b64 A load + one b64 Bt load + one WMMA per step.

__global__ __launch_bounds__(256) void k_gemm_wmma_f32(const float* __restrict__ A,
                                                       const float* __restrict__ Bt,
                                                       float* __restrict__ C,
                                                       int M, int K, int Ncols,
                                                       int ntilesN) {
  int waveId = (int)((blockIdx.x * (unsigned)blockDim.x + threadIdx.x) >> 5);
  int lane = threadIdx.x & 31;
  int mt = waveId / ntilesN;
  int nt = waveId - mt * ntilesN;
  int row0 = mt * 16;
  if (row0 >= M) return;               // wave-uniform: EXEC stays all-1s inside
  int col0 = nt * 16;
  int half = lane >> 4;                // 0: lanes 0-15, 1: lanes 16-31
  int r    = lane & 15;

  const float* arow = A + (size_t)(row0 + r) * K;       // A-frag row = M index
  const float* bcol = Bt + (size_t)(col0 + r) * K;      // B-frag col = N index (padded)

  v8f acc = {};
  #pragma unroll 4
  for (int k = 0; k < K; k += 4) {
    int ka = k + 2 * half;             // lanes 0-15: K=k,k+1 ; lanes 16-31: K=k+2,k+3
    v2f a = *(const v2f*)(arow + ka);  // global_load_b64
    v2f b = *(const v2f*)(bcol + ka);  // global_load_b64 (contiguous in transposed B)
    acc = __builtin_amdgcn_wmma_f32_16x16x4_f32(
        /*neg_a=*/false, a, /*neg_b=*/false, b,
        /*c_mod=*/(short)0, acc, /*reuse_a=*/false, /*reuse_b=*/false);
  }

  int cb = col0 + r;
  if (cb < Ncols) {                    // ragged-N masking only on stores
    #pragma unroll
    for (int v = 0; v < 8; ++v)        // C/D: VGPR v holds M = v (+8 for hi half)
      C[(size_t)(row0 + v + 8 * half) * Ncols + cb] = acc[v];
  }
}

// ---------------- aggregation ----------------

// agg[i,f] = dinv[i]^2 * h[i,f]   (self-loop term; initializes the buffer)
__global__ __launch_bounds__(256) void k_selfloop(const float* __restrict__ h,
                                                  const float* __restrict__ dinv,
                                                  float* __restrict__ agg,
                                                  int total, int F) {
  int i = blockIdx.x * blockDim.x + threadIdx.x;
  if (i >= total) return;
  int node = i / F;
  float d = dinv[node];
  agg[i] = d * d * h[i];
}

// edge scatter-add, F = 64 (shifts)
__global__ __launch_bounds__(256) void k_edge_agg64(const int* __restrict__ row,
                                                    const int* __restrict__ col,
                                                    const float* __restrict__ dinv,
                                                    const float* __restrict__ h,
                                                    float* __restrict__ agg, int E) {
  int idx = blockIdx.x * blockDim.x + threadIdx.x;  // E*64 = 204.8M < 2^31
  int e = idx >> 6;
  int f = idx & 63;
  if (e >= E) return;
  int rs = row[e], cd = col[e];
  float w = dinv[rs] * dinv[cd];
  atomicAdd(&agg[(size_t)cd * 64 + f], w * h[(size_t)rs * 64 + f]);
}

// edge scatter-add, F = 40
__global__ __launch_bounds__(256) void k_edge_agg40(const int* __restrict__ row,
                                                    const int* __restrict__ col,
                                                    const float* __restrict__ dinv,
                                                    const float* __restrict__ h,
                                                    float* __restrict__ agg, int E) {
  int idx = blockIdx.x * blockDim.x + threadIdx.x;  // E*40 = 128M < 2^31
  if (idx >= E * 40) return;
  int e = idx / 40;
  int f = idx - e * 40;
  int rs = row[e], cd = col[e];
  float w = dinv[rs] * dinv[cd];
  atomicAdd(&agg[(size_t)cd * 40 + f], w * h[(size_t)rs * 40 + f]);
}

// h[i,f] = relu(h[i,f] + b[f]), F = 64, in place
__global__ __launch_bounds__(256) void k_bias_relu64(float* __restrict__ h,
                                                     const float* __restrict__ b,
                                                     int total) {
  int i = blockIdx.x * blockDim.x + threadIdx.x;
  if (i >= total) return;
  float v = h[i] + b[i & 63];
  h[i] = v > 0.0f ? v : 0.0f;
}

// ---------------- log_softmax over 40 classes, one wave per node ----------------

__global__ __launch_bounds__(256) void k_logsoftmax40(const float* __restrict__ z,
                                                      const float* __restrict__ b,
                                                      float* __restrict__ out, int N) {
  int node = blockIdx.x * 8 + (threadIdx.x >> 5);
  int lane = threadIdx.x & 31;
  if (node >= N) return;
  const float* zr = z + (size_t)node * NCLS;
  float v0 = zr[lane] + b[lane];                     // lane 0..31 < 40
  bool has2 = (lane + 32) < NCLS;                    // lanes 0..7
  float v1 = has2 ? (zr[lane + 32] + b[lane + 32]) : -__builtin_huge_valf();

  float m = fmaxf(v0, v1);
  #pragma unroll
  for (int off = 16; off > 0; off >>= 1) m = fmaxf(m, __shfl_xor(m, off, 32));

  float s = expf(v0 - m) + (has2 ? expf(v1 - m) : 0.0f);
  #pragma unroll
  for (int off = 16; off > 0; off >>= 1) s += __shfl_xor(s, off, 32);

  float lse = m + logf(s);
  float* outr = out + (size_t)node * NCLS;
  outr[lane] = v0 - lse;
  if (has2) outr[lane + 32] = v1 - lse;
}

// ---------------- host launch ----------------

extern "C" void kernel_launch(void* const* d_in, const int* in_sizes, int n_in,
                              void* d_out, int out_size, void* d_ws, size_t ws_size,
                              hipStream_t stream) {
  const float* x  = (const float*)d_in[0];
  const int*   ei = (const int*)d_in[1];   // edge_index, shape (2, E)
  const float* W1 = (const float*)d_in[2];
  const float* b1 = (const float*)d_in[3];
  const float* W2 = (const float*)d_in[4];
  const float* b2 = (const float*)d_in[5];

  const int N = in_sizes[0] / NFEAT;       // 100000
  const int E = in_sizes[1] / 2;           // 3200000
  const int* row = ei;
  const int* col = ei + E;

  // workspace layout (floats):
  // dinv[N] | bufA[N*64] | bufB[N*64] | bufC[N*40] | Bt1[64*128] | Bt2[48*64]
  float* dinv = (float*)d_ws;
  float* bufA = dinv + N;                        // h1, later h2
  float* bufB = bufA + (size_t)N * NHID;         // agg1 -> relu'd hidden
  float* bufC = bufB + (size_t)N * NHID;         // agg2
  float* Bt1  = bufC + (size_t)N * NCLS;         // W1^T padded: [64 x 128]
  float* Bt2  = Bt1 + NHID * NFEAT;              // W2^T padded: [48 x 64]

  const int T = 256;

  // --- weight prep (tiny) ---
  k_pad_transpose<<<(NHID * NFEAT + T - 1) / T, T, 0, stream>>>(W1, Bt1, NFEAT, NHID, NHID);
  k_pad_transpose<<<(NCLS_PAD * NHID + T - 1) / T, T, 0, stream>>>(W2, Bt2, NHID, NCLS, NCLS_PAD);

  // --- normalization: deg (init 1 for self loop) -> dinv ---
  k_fill1<<<(N + T - 1) / T, T, 0, stream>>>(dinv, N);
  k_deg_accum<<<(E + T - 1) / T, T, 0, stream>>>(col, dinv, E);
  k_rsqrt<<<(N + T - 1) / T, T, 0, stream>>>(dinv, N);

  // --- layer 1: h1 = x @ W1 (WMMA fp32, branch-free inner loop) ---
  int mtiles = (N + 15) / 16;                    // 6250 (exact)
  int waves1 = mtiles * (NHID / 16);             // 25000
  k_gemm_wmma_f32<<<(waves1 * 32 + T - 1) / T, T, 0, stream>>>(
      x, Bt1, bufA, N, NFEAT, NHID, NHID / 16);

  // --- layer 1 aggregation: self loop + edges, then bias+relu ---
  int tot64 = N * NHID;
  k_selfloop<<<(tot64 + T - 1) / T, T, 0, stream>>>(bufA, dinv, bufB, tot64, NHID);
  k_edge_agg64<<<(E * 64 + T - 1) / T, T, 0, stream>>>(row, col, dinv, bufA, bufB, E);
  k_bias_relu64<<<(tot64 + T - 1) / T, T, 0, stream>>>(bufB, b1, tot64);

  // --- layer 2: h2 = h @ W2 (WMMA fp32, 48 padded cols -> masked stores) ---
  int ntiles2 = NCLS_PAD / 16;                   // 3
  int waves2 = mtiles * ntiles2;
  k_gemm_wmma_f32<<<(waves2 * 32 + T - 1) / T, T, 0, stream>>>(
      bufB, Bt2, bufA, N, NHID, NCLS, ntiles2);

  // --- layer 2 aggregation ---
  int tot40 = N * NCLS;
  k_selfloop<<<(tot40 + T - 1) / T, T, 0, stream>>>(bufA, dinv, bufC, tot40, NCLS);
  k_edge_agg40<<<(E * 40 + T - 1) / T, T, 0, stream>>>(row, col, dinv, bufA, bufC, E);

  // --- bias + log_softmax -> d_out ---
  k_logsoftmax40<<<(N + 7) / 8, T, 0, stream>>>(bufC, b2, (float*)d_out, N);
}